// MatrixFactorization_5231270167003
// MI455X (gfx1250) — compile-verified
//
#include <hip/hip_runtime.h>

#define NUSERS 32768
#define NITEMS 8192
#define DIM    128
#define BATCH  8192
#define NNZ    5000000
#define LAM1   0.1f
#define LAM2   0.1f

typedef __attribute__((ext_vector_type(2))) float v2f;
typedef __attribute__((ext_vector_type(8))) float v8f;
typedef unsigned int u32x4 __attribute__((ext_vector_type(4)));
typedef int i32x4 __attribute__((ext_vector_type(4)));
typedef int i32x8 __attribute__((ext_vector_type(8)));

// ---- workspace layout (float offsets) ----
// [0,      8192)  colsum[i]
// [8192,  16384)  base[b]   (baseline + svd score)
// [16384, 24576)  regRow[b] (per-row regularization partial)
// [32768, ...  )  sim[b][i] row-major BATCH x NITEMS
#define WS_COLSUM 0
#define WS_BASE   8192
#define WS_REG    16384
#define WS_SIM    32768

#if __has_builtin(__builtin_amdgcn_tensor_load_to_lds)
#define HAVE_TDM 1
#endif

// ---------------------------------------------------------------- helpers
__device__ __forceinline__ float keyval(unsigned int k) {
  // inverse of the order-preserving float->uint mapping
  unsigned int u = (k & 0x80000000u) ? (k ^ 0x80000000u) : ~k;
  return __uint_as_float(u);
}

__device__ __forceinline__ unsigned int lds_offset(const void* p) {
  // generic -> LDS address space; low 32 bits are the LDS byte offset
  return (unsigned int)(unsigned long long)(__attribute__((address_space(3))) const void*)p;
}

// ---------------------------------------------------------------- kernels
__global__ void zero_colsum_kernel(float* ws) {
  int t = blockIdx.x * blockDim.x + threadIdx.x;
  if (t < NITEMS) ws[WS_COLSUM + t] = 0.0f;
}

__global__ void colsum_kernel(const int* __restrict__ row_idx,
                              const int* __restrict__ col_idx,
                              const float* __restrict__ rating,
                              const float* __restrict__ uavg,
                              const float* __restrict__ mavg,
                              const float* __restrict__ gmean,
                              float* __restrict__ ws) {
  int t = blockIdx.x * blockDim.x + threadIdx.x;
  if (t >= NNZ) return;
  int c = col_idx[t];
  float r = rating[t] - (gmean[0] + uavg[row_idx[t]] + mavg[c]);
  atomicAdd(&ws[WS_COLSUM + c], r);
}

// one wave per batch row: baseline + svd dot + reg partial
__global__ __launch_bounds__(256) void base_kernel(const int* __restrict__ uidx,
                                                   const int* __restrict__ iidx,
                                                   const float* __restrict__ uemb,
                                                   const float* __restrict__ iemb,
                                                   const float* __restrict__ uavg,
                                                   const float* __restrict__ mavg,
                                                   const float* __restrict__ gmean,
                                                   float* __restrict__ ws) {
  int w = threadIdx.x >> 5, lane = threadIdx.x & 31;
  int row = blockIdx.x * 8 + w;
  if (row >= BATCH) return;
  int u = uidx[row], it = iidx[row];
  float4 uv = *(const float4*)&uemb[(size_t)u * DIM + lane * 4];
  float4 iv = *(const float4*)&iemb[(size_t)it * DIM + lane * 4];
  float dot = uv.x * iv.x + uv.y * iv.y + uv.z * iv.z + uv.w * iv.w;
  float rg = LAM1 * (uv.x * uv.x + uv.y * uv.y + uv.z * uv.z + uv.w * uv.w) +
             LAM2 * (iv.x * iv.x + iv.y * iv.y + iv.z * iv.z + iv.w * iv.w);
  for (int off = 16; off > 0; off >>= 1) {
    dot += __shfl_xor(dot, off, 32);
    rg  += __shfl_xor(rg, off, 32);
  }
  if (lane == 0) {
    ws[WS_BASE + row] = uavg[u] + mavg[it] - gmean[0] + dot;
    ws[WS_REG + row]  = rg;
  }
}

// sim[m][n] = sum_k item_emb[item_indices[m]][k] * item_emb[n][k]
// fp32 WMMA 16x16x4. Block = 256 threads = 8 waves; block tile 128(M) x 128(N);
// wave tile 32(M) x 64(N) = 2x4 subtiles of 16x16.
// Full-K panels staged in LDS at padded stride 132 floats:
//   B panel (contiguous 64KB of item_emb) via one TENSOR_LOAD_TO_LDS with
//   LDS padding (pad every 128 dwords by 4 dwords -> stride 132);
//   A panel (index-gathered rows) via global_load_b128.
#define BSTRIDE 132
#define GEMM_SMEM_FLOATS (2 * 128 * BSTRIDE + 128)

__global__ __launch_bounds__(256) void gemm_kernel(const int* __restrict__ iidx,
                                                   const float* __restrict__ iemb,
                                                   float* __restrict__ ws) {
  extern __shared__ float smem[];
  float* Abuf = smem;
  float* Bbuf = smem + 128 * BSTRIDE;
  int*   sIdx = (int*)(smem + 2 * 128 * BSTRIDE);

  float* sim = ws + WS_SIM;
  const int tid   = threadIdx.x;
  const int lane  = tid & 31;
  const int w     = tid >> 5;
  const int waveM = w >> 1;    // 0..3
  const int waveN = w & 1;     // 0..1
  const int blockM = blockIdx.y;
  const int blockN = blockIdx.x;
  const int half = lane >> 4;  // 0: lanes 0-15, 1: lanes 16-31
  const int lm   = lane & 15;

  if (tid < 128) sIdx[tid] = iidx[blockM * 128 + tid];

  const float* bPanel = iemb + (size_t)(blockN * 128) * DIM;  // 128x128 f32, contiguous

#ifdef HAVE_TDM
  if (w == 0) {
    // ---- Tensor DMA descriptor: 2D tile 128(dim0) x 128(dim1), f32,
    //      LDS padding: +4 dwords every 128 dwords -> row stride 132 floats
    unsigned long long ga = (unsigned long long)bPanel;
    u32x4 g0 = { 1u,                                   // count=1, is_restore=0, gather off
                 lds_offset(Bbuf),                     // lds_addr (bytes)
                 (unsigned int)ga,                     // global_addr[31:0]
                 (unsigned int)((ga >> 32) & 0x01FFFFFFull) | (2u << 30) };  // addr[56:32], type=2
    i32x8 g1 = { (int)((2u << 16) |                    // data_size = 4B
                       (1u << 20) |                    // pad_enable
                       (6u << 22) |                    // pad_interval: 128 dwords
                       (3u << 25)),                    // pad_amount: 4 dwords
                 (int)(128u << 16),                    // tensor_dim0[15:0] = 128
                 (int)(8192u << 16),                   // dim0[31:16]=0 | tensor_dim1[15:0] = 8192
                 (int)(128u << 16),                    // tensor_dim1[31:16]=0 | tile_dim0 = 128
                 128,                                  // tile_dim1 = 128, tile_dim2 = 0
                 128,                                  // tensor_dim0_stride = 128 elements
                 0, 0 };                               // stride hi / tensor_dim1_stride = 0
    i32x4 gz4 = { 0, 0, 0, 0 };
    i32x8 gz8 = { 0, 0, 0, 0, 0, 0, 0, 0 };
    __builtin_amdgcn_tensor_load_to_lds(g0, g1, gz4, gz4, gz8, 0);
  }
#endif
  __syncthreads();  // sIdx visible

  // stage gathered A panel: 128 rows x 128 floats (4096 float4 / 256 threads)
#pragma unroll
  for (int i = 0; i < 16; ++i) {
    int fid = tid + i * 256;   // 0..4095
    int row = fid >> 5;        // 32 float4 per row
    int c4  = fid & 31;
    float4 av = *(const float4*)&iemb[(size_t)sIdx[row] * DIM + c4 * 4];
    *(float4*)&Abuf[row * BSTRIDE + c4 * 4] = av;
  }

#ifdef HAVE_TDM
  if (w == 0) __builtin_amdgcn_s_wait_tensorcnt(0);
#else
  // fallback: stage B panel with vector loads
#pragma unroll
  for (int i = 0; i < 16; ++i) {
    int fid = tid + i * 256;
    int row = fid >> 5;
    int c4  = fid & 31;
    float4 bv = *(const float4*)&bPanel[(size_t)row * DIM + c4 * 4];
    *(float4*)&Bbuf[row * BSTRIDE + c4 * 4] = bv;
  }
#endif
  __syncthreads();

  const v8f vzero = {0.f, 0.f, 0.f, 0.f, 0.f, 0.f, 0.f, 0.f};
  v8f acc[2][4];
#pragma unroll
  for (int mi = 0; mi < 2; ++mi)
#pragma unroll
    for (int ni = 0; ni < 4; ++ni) acc[mi][ni] = vzero;

#pragma unroll
  for (int k0 = 0; k0 < DIM; k0 += 4) {
    v2f a[2], b[4];
#pragma unroll
    for (int mi = 0; mi < 2; ++mi)
      a[mi] = *(const v2f*)&Abuf[(waveM * 32 + mi * 16 + lm) * BSTRIDE + k0 + 2 * half];
#pragma unroll
    for (int ni = 0; ni < 4; ++ni)
      b[ni] = *(const v2f*)&Bbuf[(waveN * 64 + ni * 16 + lm) * BSTRIDE + k0 + 2 * half];
#pragma unroll
    for (int mi = 0; mi < 2; ++mi)
#pragma unroll
      for (int ni = 0; ni < 4; ++ni)
        acc[mi][ni] = __builtin_amdgcn_wmma_f32_16x16x4_f32(
            false, a[mi], false, b[ni], (short)0, acc[mi][ni], false, false);
  }

  // write 16x16 f32 C/D tiles: vgpr r -> M = r + 8*half, N = lane&15
#pragma unroll
  for (int mi = 0; mi < 2; ++mi)
#pragma unroll
    for (int ni = 0; ni < 4; ++ni) {
      int colg = blockN * 128 + waveN * 64 + ni * 16 + lm;
      int rowb = blockM * 128 + waveM * 32 + mi * 16 + 8 * half;
#pragma unroll
      for (int r = 0; r < 8; ++r)
        sim[(size_t)(rowb + r) * NITEMS + colg] = acc[mi][ni][r];
    }
}

// one block per batch row: radix-select K-th largest, weighted top-k sum,
// fused relu(base + contrib) -> out[row]
__global__ __launch_bounds__(256) void topk_kernel(const float* __restrict__ ws,
                                                   const int* __restrict__ topk_ptr,
                                                   float* __restrict__ out) {
  __shared__ unsigned int skey[NITEMS];
  __shared__ unsigned int hist[256];
  __shared__ unsigned int sscan[256];
  __shared__ float sred[256];
  __shared__ unsigned int sPref, sRemK, sRun;

  const int tid = threadIdx.x;
  const int q = blockIdx.x;
  const float* sim = ws + WS_SIM + (size_t)q * NITEMS;
  const float* colsum = ws + WS_COLSUM;

  // order-preserving float -> uint keys
#pragma unroll 4
  for (int i = 0; i < NITEMS / 256; ++i) {
    int j = tid + i * 256;
    unsigned int u = __float_as_uint(sim[j]);
    skey[j] = (u & 0x80000000u) ? ~u : (u | 0x80000000u);
  }
  if (tid == 0) { sPref = 0u; sRemK = (unsigned int)topk_ptr[0]; }
  __syncthreads();

  // 4-pass MSB radix select for the K-th largest key
  for (int pass = 0; pass < 4; ++pass) {
    int p = 24 - pass * 8;
    hist[tid] = 0u;
    __syncthreads();
    unsigned int pref = sPref;
    unsigned int mask = (pass == 0) ? 0u : (0xFFFFFFFFu << (p + 8));
    for (int i = 0; i < NITEMS / 256; ++i) {
      unsigned int k = skey[tid + i * 256];
      if ((k & mask) == (pref & mask)) atomicAdd(&hist[(k >> p) & 255u], 1u);
    }
    __syncthreads();
    if (tid == 0) {
      unsigned int remK = sRemK, cum = 0u;
      for (int b = 255; b >= 0; --b) {
        unsigned int c = hist[b];
        if (cum + c >= remK) {
          sPref = pref | ((unsigned int)b << p);
          sRemK = remK - cum;
          break;
        }
        cum += c;
      }
    }
    __syncthreads();
  }
  const unsigned int T = sPref;     // K-th largest key
  const unsigned int need = sRemK;  // how many == T still belong to the top-k

  // strictly-greater contributions
  float accv = 0.0f;
  for (int i = 0; i < NITEMS / 256; ++i) {
    int j = tid + i * 256;
    unsigned int k = skey[j];
    if (k > T) accv += keyval(k) * colsum[j];
  }

  // tie pass: take `need` elements equal to T in ascending index order
  if (tid == 0) sRun = 0u;
  __syncthreads();
  for (int c = 0; c < NITEMS / 256; ++c) {
    if (sRun >= need) break;  // uniform: shared value read after a barrier
    int j = c * 256 + tid;
    unsigned int flag = (skey[j] == T) ? 1u : 0u;
    sscan[tid] = flag;
    __syncthreads();
    for (int off = 1; off < 256; off <<= 1) {
      unsigned int v = (tid >= off) ? sscan[tid - off] : 0u;
      __syncthreads();
      sscan[tid] += v;
      __syncthreads();
    }
    unsigned int rank = sRun + sscan[tid] - flag;  // exclusive rank among ties
    if (flag && rank < need) accv += keyval(skey[j]) * colsum[j];
    unsigned int tot = sscan[255];
    __syncthreads();
    if (tid == 0) sRun += tot;
    __syncthreads();
  }

  // block reduction + fused epilogue
  sred[tid] = accv;
  __syncthreads();
  for (int off = 128; off > 0; off >>= 1) {
    if (tid < off) sred[tid] += sred[tid + off];
    __syncthreads();
  }
  if (tid == 0) {
    float pred = ws[WS_BASE + q] + sred[0];
    out[q] = fmaxf(pred, 0.0f);
  }
}

__global__ void reg_reduce_kernel(const float* __restrict__ ws, float* __restrict__ out) {
  __shared__ float s[256];
  int tid = threadIdx.x;
  float a = 0.0f;
  for (int i = tid; i < BATCH; i += 256) a += ws[WS_REG + i];
  s[tid] = a;
  __syncthreads();
  for (int off = 128; off > 0; off >>= 1) {
    if (tid < off) s[tid] += s[tid + off];
    __syncthreads();
  }
  if (tid == 0) out[BATCH] = s[0];
}

// ---------------------------------------------------------------- launch
extern "C" void kernel_launch(void* const* d_in, const int* in_sizes, int n_in,
                              void* d_out, int out_size, void* d_ws, size_t ws_size,
                              hipStream_t stream) {
  (void)in_sizes; (void)n_in; (void)out_size; (void)ws_size;
  const int*   user_indices = (const int*)d_in[0];
  const int*   item_indices = (const int*)d_in[1];
  const int*   row_idx      = (const int*)d_in[2];
  const int*   col_idx      = (const int*)d_in[3];
  const float* rating       = (const float*)d_in[4];
  const float* uemb         = (const float*)d_in[5];
  const float* iemb         = (const float*)d_in[6];
  const float* uavg         = (const float*)d_in[7];
  const float* mavg         = (const float*)d_in[8];
  const float* gmean        = (const float*)d_in[9];
  const int*   topk         = (const int*)d_in[10];
  float* ws  = (float*)d_ws;
  float* out = (float*)d_out;

  hipLaunchKernelGGL(zero_colsum_kernel, dim3((NITEMS + 255) / 256), dim3(256), 0, stream, ws);
  hipLaunchKernelGGL(colsum_kernel, dim3((NNZ + 255) / 256), dim3(256), 0, stream,
                     row_idx, col_idx, rating, uavg, mavg, gmean, ws);
  hipLaunchKernelGGL(base_kernel, dim3(BATCH / 8), dim3(256), 0, stream,
                     user_indices, item_indices, uemb, iemb, uavg, mavg, gmean, ws);
  hipLaunchKernelGGL(gemm_kernel, dim3(NITEMS / 128, BATCH / 128), dim3(256),
                     GEMM_SMEM_FLOATS * sizeof(float), stream, item_indices, iemb, ws);
  hipLaunchKernelGGL(topk_kernel, dim3(BATCH), dim3(256), 0, stream, ws, topk, out);
  hipLaunchKernelGGL(reg_reduce_kernel, dim3(1), dim3(256), 0, stream, ws, out);
}